// GINE_Allocation_Predictor_49993419325800
// MI455X (gfx1250) — compile-verified
//
#include <hip/hip_runtime.h>
#include <hip/hip_bf16.h>

typedef __attribute__((ext_vector_type(2))) float v2f;
typedef __attribute__((ext_vector_type(8))) float v8f;

// ---------------------------------------------------------------------------
// WMMA helper: D(16x16) += A(16 x K) * W(K x 16) using V_WMMA_F32_16X16X4_F32.
// A is LDS row-major (stride lda), W is LDS row-major (stride ldw).
// A-fragment layout (32-bit A 16x4): lanes 0-15 hold K=k,k+1; lanes 16-31 hold
// K=k+2,k+3. B-fragment: vgpr0 = rows k+hi, vgpr1 = rows k+2+hi.
// ---------------------------------------------------------------------------
__device__ __forceinline__ v8f wmma_f32_tile(const float* __restrict__ As, int lda,
                                             const float* __restrict__ Ws, int ldw,
                                             int m0, int n0, int K, int lane) {
    const int hi = (lane >> 4) & 1;
    const int lm = lane & 15;
    v8f acc = {};
    for (int k = 0; k < K; k += 4) {
        v2f a, b;
        const float* ar = As + (m0 + lm) * lda + k + 2 * hi;
        a.x = ar[0];
        a.y = ar[1];
        b.x = Ws[(k + hi) * ldw + (n0 + lm)];
        b.y = Ws[(k + 2 + hi) * ldw + (n0 + lm)];
        acc = __builtin_amdgcn_wmma_f32_16x16x4_f32(false, a, false, b,
                                                    (short)0, acc, false, false);
    }
    return acc;
}

// ---------------------------------------------------------------------------
// Misc small kernels
// ---------------------------------------------------------------------------
__global__ void init_copy_kernel(const float* __restrict__ src, float* __restrict__ dst, int n) {
    int i = blockIdx.x * blockDim.x + threadIdx.x;
    if (i < n) dst[i] = src[i];
}

__global__ void init_seg_kernel(float* seg_m, float* seg_s, int g) {
    int i = blockIdx.x * blockDim.x + threadIdx.x;
    if (i < g) {
        seg_m[i] = __int_as_float(0xFF800000);  // -inf
        seg_s[i] = 0.0f;
    }
}

// ---------------------------------------------------------------------------
// conv1 edge pass: agg1[dst] += relu(x[src] + ea*we1 + be1), F=9, thread/edge
// ---------------------------------------------------------------------------
__global__ void conv1_edge_kernel(const float* __restrict__ x,
                                  const int* __restrict__ src, const int* __restrict__ dst,
                                  const float* __restrict__ ea,
                                  const float* __restrict__ we1, const float* __restrict__ be1,
                                  float* __restrict__ agg1, int E) {
    int e = blockIdx.x * blockDim.x + threadIdx.x;
    if (e >= E) return;
    int s = src[e], d = dst[e];
    float a = ea[e];
    const float* xs = x + (size_t)s * 9;
    float* od = agg1 + (size_t)d * 9;
#pragma unroll
    for (int j = 0; j < 9; ++j) {
        float m = fmaxf(fmaf(a, we1[j], be1[j]) + xs[j], 0.0f);
        atomicAdd(&od[j], m);
    }
}

// ---------------------------------------------------------------------------
// MLP1: hin[N,9] (= x+agg) -> relu(@w11+b11) -> relu(@w12+b12) -> h1, agg2
// 64-node tile per block, 256 threads (8 waves). Layer2 uses WMMA.
// ---------------------------------------------------------------------------
__global__ void mlp1_kernel(const float* __restrict__ hin,
                            const float* __restrict__ w11, const float* __restrict__ b11,
                            const float* __restrict__ w12, const float* __restrict__ b12,
                            float* __restrict__ h1, float* __restrict__ agg2, int N) {
    __shared__ float s_in[64 * 9];
    __shared__ float s_w11[9 * 64];
    __shared__ float s_t1[64 * 65];   // padded stride: conflict-free A-frag reads
    __shared__ float s_w12[64 * 64];

    const int base = blockIdx.x * 64;
    const int t = threadIdx.x;

    for (int i = t; i < 64 * 9; i += 256) {
        int r = i / 9, c = i - r * 9;
        int node = base + r;
        s_in[i] = (node < N) ? hin[(size_t)node * 9 + c] : 0.0f;
    }
    for (int i = t; i < 9 * 64; i += 256) s_w11[i] = w11[i];
    for (int i = t; i < 64 * 64; i += 256) s_w12[i] = w12[i];
    __syncthreads();

    // layer 1 (K=9): thread t -> node (t&63), feature block ((t>>6)*16 .. +16)
    {
        const int r = t & 63;
        const int f0 = (t >> 6) * 16;
#pragma unroll
        for (int f = f0; f < f0 + 16; ++f) {
            float acc = b11[f];
#pragma unroll
            for (int k = 0; k < 9; ++k) acc = fmaf(s_in[r * 9 + k], s_w11[k * 64 + f], acc);
            s_t1[r * 65 + f] = fmaxf(acc, 0.0f);
        }
    }
    __syncthreads();

    // layer 2 (64x64) via WMMA: 16 subtiles, 8 waves, 2 each
    const int wave = t >> 5;
    const int lane = t & 31;
    const int hi = (lane >> 4) & 1;
    const int lm = lane & 15;
    for (int s2 = wave; s2 < 16; s2 += 8) {
        const int m0 = (s2 >> 2) * 16, n0 = (s2 & 3) * 16;
        v8f acc = wmma_f32_tile(s_t1, 65, s_w12, 64, m0, n0, 64, lane);
        const float bias = b12[n0 + lm];
#pragma unroll
        for (int v = 0; v < 8; ++v) {
            const int rr = m0 + v + 8 * hi;
            const int node = base + rr;
            if (node < N) {
                float val = fmaxf(acc[v] + bias, 0.0f);
                size_t idx = (size_t)node * 64 + n0 + lm;
                h1[idx] = val;
                agg2[idx] = val;   // agg2 pre-seeded with h1 (GINE residual)
            }
        }
    }
}

// ---------------------------------------------------------------------------
// conv2 edge pass: 16 lanes/edge, 4 feats/lane. Coalesced b128 row gathers,
// 4 f32 atomics per lane into L2-resident agg2.
// ---------------------------------------------------------------------------
__global__ void conv2_edge_kernel(const float* __restrict__ h1,
                                  const int* __restrict__ src, const int* __restrict__ dst,
                                  const float* __restrict__ ea,
                                  const float* __restrict__ we2, const float* __restrict__ be2,
                                  float* __restrict__ agg2, int E) {
    int gid = blockIdx.x * blockDim.x + threadIdx.x;
    int e = gid >> 4;
    int l = gid & 15;
    if (e >= E) return;
    int s = src[e], d = dst[e];
    float a = ea[e];
    float4 hv = ((const float4*)(h1 + (size_t)s * 64))[l];
    float4 wv = ((const float4*)we2)[l];
    float4 bv = ((const float4*)be2)[l];
    float m0 = fmaxf(hv.x + fmaf(a, wv.x, bv.x), 0.0f);
    float m1 = fmaxf(hv.y + fmaf(a, wv.y, bv.y), 0.0f);
    float m2 = fmaxf(hv.z + fmaf(a, wv.z, bv.z), 0.0f);
    float m3 = fmaxf(hv.w + fmaf(a, wv.w, bv.w), 0.0f);
    float* od = agg2 + (size_t)d * 64 + l * 4;
    atomicAdd(od + 0, m0);
    atomicAdd(od + 1, m1);
    atomicAdd(od + 2, m2);
    atomicAdd(od + 3, m3);
}

// ---------------------------------------------------------------------------
// MLP2 + readout: agg2 -> relu(@w21) -> relu(@w22) -> relu(@wr1) -> @wr2 -> raw
// All staged in LDS, three WMMA GEMMs, 64-node tile per block.
// ---------------------------------------------------------------------------
__global__ void mlp2_kernel(const float* __restrict__ hin,
                            const float* __restrict__ w21, const float* __restrict__ b21,
                            const float* __restrict__ w22, const float* __restrict__ b22,
                            const float* __restrict__ wr1, const float* __restrict__ br1,
                            const float* __restrict__ wr2, const float* __restrict__ br2,
                            float* __restrict__ raw, int N) {
    __shared__ float tA[64 * 65];
    __shared__ float tB[64 * 65];
    __shared__ float sW[64 * 64];

    const int base = blockIdx.x * 64;
    const int t = threadIdx.x;
    const int wave = t >> 5;
    const int lane = t & 31;
    const int hi = (lane >> 4) & 1;
    const int lm = lane & 15;

    for (int i = t; i < 64 * 64; i += 256) {
        int r = i >> 6, c = i & 63;
        int node = base + r;
        tA[r * 65 + c] = (node < N) ? hin[(size_t)node * 64 + c] : 0.0f;
        sW[i] = w21[i];
    }
    __syncthreads();

    // GEMM1: tB = relu(tA @ w21 + b21)
    for (int s2 = wave; s2 < 16; s2 += 8) {
        const int m0 = (s2 >> 2) * 16, n0 = (s2 & 3) * 16;
        v8f acc = wmma_f32_tile(tA, 65, sW, 64, m0, n0, 64, lane);
        const float bias = b21[n0 + lm];
#pragma unroll
        for (int v = 0; v < 8; ++v)
            tB[(m0 + v + 8 * hi) * 65 + n0 + lm] = fmaxf(acc[v] + bias, 0.0f);
    }
    __syncthreads();
    for (int i = t; i < 64 * 64; i += 256) sW[i] = w22[i];
    __syncthreads();

    // GEMM2: tA = relu(tB @ w22 + b22)
    for (int s2 = wave; s2 < 16; s2 += 8) {
        const int m0 = (s2 >> 2) * 16, n0 = (s2 & 3) * 16;
        v8f acc = wmma_f32_tile(tB, 65, sW, 64, m0, n0, 64, lane);
        const float bias = b22[n0 + lm];
#pragma unroll
        for (int v = 0; v < 8; ++v)
            tA[(m0 + v + 8 * hi) * 65 + n0 + lm] = fmaxf(acc[v] + bias, 0.0f);
    }
    __syncthreads();
    for (int i = t; i < 64 * 32; i += 256) sW[i] = wr1[i];
    __syncthreads();

    // GEMM3: tB[:, 0:32] = relu(tA @ wr1 + br1); 8 subtiles (4M x 2N), 1/wave
    {
        const int m0 = (wave >> 1) * 16, n0 = (wave & 1) * 16;
        v8f acc = wmma_f32_tile(tA, 65, sW, 32, m0, n0, 64, lane);
        const float bias = br1[n0 + lm];
#pragma unroll
        for (int v = 0; v < 8; ++v)
            tB[(m0 + v + 8 * hi) * 65 + n0 + lm] = fmaxf(acc[v] + bias, 0.0f);
    }
    __syncthreads();

    // readout: raw[node] = dot(tB[node, 0:32], wr2) + br2
    if (t < 64) {
        int node = base + t;
        if (node < N) {
            float acc = br2[0];
#pragma unroll
            for (int k = 0; k < 32; ++k) acc = fmaf(tB[t * 65 + k], wr2[k], acc);
            raw[node] = acc;
        }
    }
}

// ---------------------------------------------------------------------------
// Segment softmax (sign-split float atomic max trick, then exp/sum, normalize)
// ---------------------------------------------------------------------------
__device__ __forceinline__ void atomicMaxFloat(float* addr, float value) {
    if (value >= 0.0f) atomicMax((int*)addr, __float_as_int(value));
    else               atomicMin((unsigned int*)addr, __float_as_uint(value));
}

__global__ void seg_max_kernel(const float* __restrict__ raw, const int* __restrict__ batch,
                               float* __restrict__ seg_m, int N) {
    int i = blockIdx.x * blockDim.x + threadIdx.x;
    if (i < N) atomicMaxFloat(&seg_m[batch[i]], raw[i]);
}

__global__ void seg_expsum_kernel(const float* __restrict__ raw, const int* __restrict__ batch,
                                  const float* __restrict__ seg_m,
                                  float* __restrict__ z, float* __restrict__ seg_s, int N) {
    int i = blockIdx.x * blockDim.x + threadIdx.x;
    if (i < N) {
        int b = batch[i];
        float e = __expf(raw[i] - seg_m[b]);
        z[i] = e;
        atomicAdd(&seg_s[b], e);
    }
}

__global__ void finalize_kernel(const float* __restrict__ z, const int* __restrict__ batch,
                                const float* __restrict__ seg_s, const float* __restrict__ B_total,
                                float* __restrict__ out, int N) {
    int i = blockIdx.x * blockDim.x + threadIdx.x;
    if (i < N) {
        int b = batch[i];
        out[i] = z[i] / seg_s[b] * B_total[b];
    }
}

// ---------------------------------------------------------------------------
// Launch
// ---------------------------------------------------------------------------
extern "C" void kernel_launch(void* const* d_in, const int* in_sizes, int n_in,
                              void* d_out, int out_size, void* d_ws, size_t ws_size,
                              hipStream_t stream) {
    const float* x        = (const float*)d_in[0];
    const int*   ei       = (const int*)  d_in[1];
    const float* eattr    = (const float*)d_in[2];
    const int*   batch    = (const int*)  d_in[3];
    const float* B_total  = (const float*)d_in[4];
    const float* we1      = (const float*)d_in[5];
    const float* be1      = (const float*)d_in[6];
    const float* w11      = (const float*)d_in[7];
    const float* b11      = (const float*)d_in[8];
    const float* w12      = (const float*)d_in[9];
    const float* b12      = (const float*)d_in[10];
    const float* we2      = (const float*)d_in[11];
    const float* be2      = (const float*)d_in[12];
    const float* w21      = (const float*)d_in[13];
    const float* b21      = (const float*)d_in[14];
    const float* w22      = (const float*)d_in[15];
    const float* b22      = (const float*)d_in[16];
    const float* wr1      = (const float*)d_in[17];
    const float* br1      = (const float*)d_in[18];
    const float* wr2      = (const float*)d_in[19];
    const float* br2      = (const float*)d_in[20];

    const int N = in_sizes[0] / 9;
    const int E = in_sizes[1] / 2;
    const int G = in_sizes[4];
    const int* src = ei;
    const int* dst = ei + E;

    // carve workspace
    char* ws = (char*)d_ws;
    auto carve = [&](size_t bytes) -> char* {
        char* p = ws;
        ws += (bytes + 255) & ~(size_t)255;
        return p;
    };
    float* agg1  = (float*)carve((size_t)N * 9 * sizeof(float));
    float* h1    = (float*)carve((size_t)N * 64 * sizeof(float));
    float* agg2  = (float*)carve((size_t)N * 64 * sizeof(float));
    float* raw   = (float*)carve((size_t)N * sizeof(float));
    float* zbuf  = (float*)carve((size_t)N * sizeof(float));
    float* seg_m = (float*)carve((size_t)G * sizeof(float));
    float* seg_s = (float*)carve((size_t)G * sizeof(float));
    (void)ws_size; (void)n_in; (void)out_size;

    const int B = 256;
    const int tiles = (N + 63) / 64;

    init_copy_kernel<<<(N * 9 + B - 1) / B, B, 0, stream>>>(x, agg1, N * 9);
    init_seg_kernel<<<(G + B - 1) / B, B, 0, stream>>>(seg_m, seg_s, G);

    conv1_edge_kernel<<<(E + B - 1) / B, B, 0, stream>>>(x, src, dst, eattr, we1, be1, agg1, E);
    mlp1_kernel<<<tiles, B, 0, stream>>>(agg1, w11, b11, w12, b12, h1, agg2, N);

    long long conv2_threads = (long long)E * 16;
    conv2_edge_kernel<<<(unsigned)((conv2_threads + B - 1) / B), B, 0, stream>>>(
        h1, src, dst, eattr, we2, be2, agg2, E);

    mlp2_kernel<<<tiles, B, 0, stream>>>(agg2, w21, b21, w22, b22, wr1, br1, wr2, br2, raw, N);

    seg_max_kernel<<<(N + B - 1) / B, B, 0, stream>>>(raw, batch, seg_m, N);
    seg_expsum_kernel<<<(N + B - 1) / B, B, 0, stream>>>(raw, batch, seg_m, zbuf, seg_s, N);
    finalize_kernel<<<(N + B - 1) / B, B, 0, stream>>>(zbuf, batch, seg_s, B_total,
                                                      (float*)d_out, N);
}